// NPSCalculator_5299989643614
// MI455X (gfx1250) — compile-verified
//
#include <hip/hip_runtime.h>

typedef __attribute__((ext_vector_type(2))) float v2f;
typedef __attribute__((ext_vector_type(4))) float f4;
typedef __attribute__((ext_vector_type(8))) float v8f;

#define NUM_COLORS 30
#define BATCH      8
#define HW         (512 * 512)
#define NPIX       (BATCH * HW)          // 2,097,152 pixels (b,h,w)
#define TOTAL_SIZE (BATCH * 3 * HW)      // 6,291,456 elements (incl. channel dim)
#define CHUNK      64                    // pixels per wave-iteration
#define NCHUNKS    (NPIX / CHUNK)        // 32,768

__global__ void nps_init(float* out) { out[0] = 0.0f; }

// Cross-half (lane l <-> l^16) swap entirely in the VALU pipe: v_permlanex16
// with identity nibble selects. Avoids the ds_bpermute + s_wait_dscnt that
// __shfl_xor(.,16) costs in the inner loop.
__device__ __forceinline__ float xhalf_swap(float x) {
    int xi = __builtin_bit_cast(int, x);
    int r  = __builtin_amdgcn_permlanex16(xi, xi, 0x76543210, 0xfedcba98,
                                          /*fi=*/false, /*bound_ctrl=*/false);
    return __builtin_bit_cast(float, r);
}

__global__ __launch_bounds__(256) void nps_kernel(const float* __restrict__ patch,
                                                  const float* __restrict__ prn,
                                                  float* __restrict__ out)
{
    const int lane = threadIdx.x & 31;
    const int half = lane >> 4;   // 0: lanes 0-15, 1: lanes 16-31
    const int mrow = lane & 15;

    // ---- Build constant WMMA operands: A = colors, C = |c'|^2 + 1e-6 ----
    // c' = c - 1e-6 folds the reference's per-channel +1e-6 into the color.
    auto cprime = [&](int m, int ch) -> float {
        return (m < NUM_COLORS) ? (prn[m * 3 + ch] - 1e-6f) : 0.0f;
    };
    // A (16x4 f32): v0 = {K=0 | K=2}, v1 = {K=1 | K=3}; M = lane%16 = color.
    v2f A1, A2;
    {
        const int m1 = mrow, m2 = mrow + 16;
        A1.x = half ? cprime(m1, 2) : cprime(m1, 0);
        A1.y = half ? 1.0f          : cprime(m1, 1);
        A2.x = half ? cprime(m2, 2) : cprime(m2, 0);
        A2.y = half ? 1.0f          : cprime(m2, 1);
    }
    // C (16x16 f32): VGPR v holds M=v (lanes 0-15) / M=v+8 (lanes 16-31).
    v8f C1, C2;
    #pragma unroll
    for (int v = 0; v < 8; ++v) {
        const int m1 = v + (half ? 8 : 0);
        const int m2 = m1 + 16;
        float a = cprime(m1, 0), b = cprime(m1, 1), c = cprime(m1, 2);
        C1[v] = a * a + b * b + c * c + 1e-6f;
        float r2 = 1e30f;                           // padded colors never win the min
        if (m2 < NUM_COLORS) {
            a = cprime(m2, 0); b = cprime(m2, 1); c = cprime(m2, 2);
            r2 = a * a + b * b + c * c + 1e-6f;
        }
        C2[v] = r2;
    }

    const int wave   = blockIdx.x * (blockDim.x >> 5) + (threadIdx.x >> 5);
    const int nwaves = gridDim.x * (blockDim.x >> 5);

    float acc = 0.0f;

    for (int g = wave; g < NCHUNKS; g += nwaves) {
        const int t0 = g * CHUNK;         // chunk stays inside one batch image (HW % 64 == 0)
        const int b  = t0 / HW;
        const int hw = t0 - b * HW;
        const float* base = patch + (size_t)b * 3 * HW + hw + mrow * 4;
        const f4 p0 = *(const f4*)(base);
        const f4 p1 = *(const f4*)(base + HW);
        const f4 p2 = *(const f4*)(base + 2 * HW);

        // Prefetch next grid-stride chunk's three channel lines (global_prefetch_b8).
        const int gn = g + nwaves;
        if (gn < NCHUNKS) {
            const int tn  = gn * CHUNK;
            const int bn  = tn / HW;
            const int hwn = tn - bn * HW;
            const float* nb = patch + (size_t)bn * 3 * HW + hwn + mrow * 4;
            __builtin_prefetch(nb, 0, 0);
            __builtin_prefetch(nb + HW, 0, 0);
            __builtin_prefetch(nb + 2 * HW, 0, 0);
        }

        #pragma unroll
        for (int j = 0; j < 4; ++j) {
            const float a = p0[j], bb = p1[j], c = p2[j];
            const float q = fmaf(a, a, fmaf(bb, bb, c * c));   // |p|^2
            // B (4x16 f32): v0 = {K=0 | K=2}, v1 = {K=1 | K=3}; N = lane%16 = pixel.
            v2f Bv;
            Bv.x = half ? (-2.0f * c) : (-2.0f * a);
            Bv.y = half ? q           : (-2.0f * bb);

            // d^2[color, pixel] = A x B + C  (colors 0-15, then 16-31)
            v8f d1 = __builtin_amdgcn_wmma_f32_16x16x4_f32(false, A1, false, Bv,
                                                           (short)0, C1, false, false);
            v8f d2 = __builtin_amdgcn_wmma_f32_16x16x4_f32(false, A2, false, Bv,
                                                           (short)0, C2, false, false);

            // min over colors: pure VALU over the accumulator VGPRs ...
            float m = d1[0];
            #pragma unroll
            for (int v = 1; v < 8; ++v) m = fminf(m, d1[v]);
            #pragma unroll
            for (int v = 0; v < 8; ++v) m = fminf(m, d2[v]);
            // ... plus one VALU cross-half combine (colors 0-7/16-23 vs 8-15/24-31).
            m = fminf(m, xhalf_swap(m));
            // sqrt after min (monotone): 1 raw v_sqrt_f32 / pixel, args in [1e-6, 1e30].
            acc += __builtin_amdgcn_sqrtf(m);
        }
    }

    // Each pixel's sqrt was accumulated in both lane halves -> 0.5; then mean.
    acc *= 0.5f / (float)TOTAL_SIZE;
    #pragma unroll
    for (int off = 16; off >= 1; off >>= 1) acc += __shfl_xor(acc, off, 32);
    if (lane == 0) atomicAdd(out, acc);
}

extern "C" void kernel_launch(void* const* d_in, const int* in_sizes, int n_in,
                              void* d_out, int out_size, void* d_ws, size_t ws_size,
                              hipStream_t stream)
{
    const float* patch = (const float*)d_in[0];   // (8,3,512,512) f32
    const float* prn   = (const float*)d_in[1];   // (30,3) f32
    float* out = (float*)d_out;

    nps_init<<<1, 1, 0, stream>>>(out);
    // 1024 blocks x 8 waves = 8192 waves; 64 pixels/wave-iter -> 4 grid-stride iters.
    nps_kernel<<<1024, 256, 0, stream>>>(patch, prn, out);
}